// imaging_layer_23837068493439
// MI455X (gfx1250) — compile-verified
//
#include <hip/hip_runtime.h>
#include <hip/hip_bf16.h>
#include <math.h>

// ---------------------------------------------------------------------------
// Problem constants (match the JAX reference)
// ---------------------------------------------------------------------------
#define MESH   2048
#define SQSZ   102
#define NBATCH 32
#define NSQ    10
#define NSEG   100
#define LASTROW 2047            // colsum covers rows [0, 2046]

// Detector square origins: START_*_PCTG * 102 + 1024 (exact integers)
__constant__ int SXC[NSQ] = { 769,  769,  769,  973,  973,  973,  973, 1177, 1177, 1177};
__constant__ int SYC[NSQ] = { 769,  973, 1177,  667,  871, 1075, 1279,  769,  973, 1177};

typedef float v2f __attribute__((ext_vector_type(2)));
typedef float v8f __attribute__((ext_vector_type(8)));

__device__ __forceinline__ bool in_rng(int x, int lo) {
    return (unsigned)(x - lo) < (unsigned)SQSZ;
}

// Branchless wiped intensity for element (r, col) of batch-base `base`.
// Returns 0 for wiped (detector-square) elements and for r >= 2047.
__device__ __forceinline__ float wiped_intensity(const float* __restrict__ ur,
                                                 const float* __restrict__ ui,
                                                 long long base, int r, int col,
                                                 bool colA, bool colB) {
    int rc = min(r, LASTROW - 1);                       // clamp -> always-valid load
    long long idx = base + (long long)rc * MESH + col;
    float a = ur[idx];
    float c = ui[idx];
    float inten = fmaf(a, a, c * c);
    // Row bands: A/C (769,1177) pair with one column set, B (973) with the other.
    bool rowAC = in_rng(r, 769) || in_rng(r, 1177);
    bool rowB  = in_rng(r, 973);
    bool wipe  = (rowAC && colA) || (rowB && colB) || (r >= LASTROW);
    return wipe ? 0.0f : inten;
}

// ---------------------------------------------------------------------------
// Kernel 1: streaming column-sum of wiped |u|^2 using WMMA as the reducer.
// Each wave owns 16 columns; V_WMMA_F32_16X16X4_F32 with an all-ones A-matrix
// sums 4 rows x 16 cols per issue into a persistent f32 accumulator (C==D).
// grid = (32 colgroup-quads, 8 row-splits, 32 batches), block = 128 (4 waves).
// ---------------------------------------------------------------------------
__global__ __launch_bounds__(128)
void colsum_wmma_kernel(const float* __restrict__ ur,
                        const float* __restrict__ ui,
                        float* __restrict__ strips /* [NBATCH][NSEG] */) {
    const int lane     = threadIdx.x & 31;
    const int wave     = threadIdx.x >> 5;
    const int colgroup = blockIdx.x * 4 + wave;     // 0..127
    const int col      = colgroup * 16 + (lane & 15);
    const int klo      = lane >> 4;                 // 0 for lanes 0-15, 1 for 16-31
    const int b        = blockIdx.z;
    const int r0start  = blockIdx.y * 256;          // 8 splits x 256 rows >= 2047

    // Per-column wipe membership (hoisted out of the row loop).
    const bool colA = in_rng(col, 769) || in_rng(col, 973) || in_rng(col, 1177);
    const bool colB = in_rng(col, 667) || in_rng(col, 871) ||
                      in_rng(col, 1075) || in_rng(col, 1279);

    const long long base = (long long)b * (long long)MESH * (long long)MESH;

    v2f amat = {1.0f, 1.0f};   // all-ones 16x4 A-matrix -> D[m,n] = sum_k B[k,n]
    v8f acc  = {};             // 16x16 f32 accumulator (all rows identical)

    #pragma unroll 4
    for (int r0 = r0start; r0 < r0start + 256; r0 += 4) {
        int ra = r0 + klo;         // this lane's first K-slot row
        int rb = ra + 2;           // this lane's second K-slot row
        v2f bm;
        bm.x = wiped_intensity(ur, ui, base, ra, col, colA, colB);
        bm.y = wiped_intensity(ur, ui, base, rb, col, colA, colB);
        // D = A(ones) x B + C : reduces 4 rows of 16 columns per instruction.
        acc = __builtin_amdgcn_wmma_f32_16x16x4_f32(
                  /*neg_a=*/false, amat, /*neg_b=*/false, bm,
                  /*c_mod=*/(short)0, acc, /*reuse_a=*/false, /*reuse_b=*/false);
    }

    // Row M=0 of D holds the column sums; every lane's acc[0] is the sum for
    // column col (lanes 16-31 duplicate via M=8). Emit once per column.
    float cs = acc[0];
    if (lane < 16) {
        int seg = (int)((col + 0.5f) * (100.0f / 2048.0f));  // exact segment map
        atomicAdd(&strips[b * NSEG + seg], cs);
    }
}

// ---------------------------------------------------------------------------
// Kernel 2: raw |u|^2 sums over the 10 detector squares (tiny re-read).
// grid = (10 squares, 32 batches), block = 256.
// ---------------------------------------------------------------------------
__global__ __launch_bounds__(256)
void sqsum_kernel(const float* __restrict__ ur,
                  const float* __restrict__ ui,
                  float* __restrict__ sq /* [NBATCH][NSQ] */) {
    const int q = blockIdx.x;
    const int b = blockIdx.y;
    const int sx = SXC[q], sy = SYC[q];
    const long long base = (long long)b * (long long)MESH * (long long)MESH;

    float acc = 0.0f;
    for (int i = threadIdx.x; i < SQSZ * SQSZ; i += 256) {
        int r = sx + i / SQSZ;
        int c = sy + i % SQSZ;
        long long idx = base + (long long)r * MESH + c;
        float a = ur[idx];
        float d = ui[idx];
        acc = fmaf(a, a, fmaf(d, d, acc));
    }

    __shared__ float red[256];
    red[threadIdx.x] = acc;
    __syncthreads();
    for (int s = 128; s > 0; s >>= 1) {
        if (threadIdx.x < s) red[threadIdx.x] += red[threadIdx.x + s];
        __syncthreads();
    }
    if (threadIdx.x == 0) sq[b * NSQ + q] = red[0];
}

// ---------------------------------------------------------------------------
// Kernel 3: concat -> max-normalize -> argmax(first 10). One block per batch.
// Output layout: [32][110] floats, then 32 labels (stored as floats).
// ---------------------------------------------------------------------------
__global__ __launch_bounds__(128)
void finalize_kernel(const float* __restrict__ sq,
                     const float* __restrict__ strips,
                     float* __restrict__ out) {
    const int b = blockIdx.x;
    const int t = threadIdx.x;

    __shared__ float vals[NSQ + NSEG];
    __shared__ float red[128];

    float v = 0.0f;
    if (t < NSQ)              v = sq[b * NSQ + t];
    else if (t < NSQ + NSEG)  v = 0.8f * strips[b * NSEG + (t - NSQ)];
    if (t < NSQ + NSEG) vals[t] = v;
    red[t] = (t < NSQ + NSEG) ? v : -INFINITY;
    __syncthreads();

    for (int s = 64; s > 0; s >>= 1) {
        if (t < s) red[t] = fmaxf(red[t], red[t + s]);
        __syncthreads();
    }
    float mx = red[0];

    if (t < NSQ + NSEG) out[b * (NSQ + NSEG) + t] = v / mx;

    if (t == 0) {
        int best = 0;
        float bv = vals[0];
        #pragma unroll
        for (int i = 1; i < NSQ; i++) {
            if (vals[i] > bv) { bv = vals[i]; best = i; }
        }
        out[NBATCH * (NSQ + NSEG) + b] = (float)best;  // labels after the 32x110 block
    }
}

// ---------------------------------------------------------------------------
// Workspace zero-init (graph-safe; harness does not re-poison between replays)
// ---------------------------------------------------------------------------
__global__ void zero_ws_kernel(float* __restrict__ ws, int n) {
    int i = blockIdx.x * blockDim.x + threadIdx.x;
    if (i < n) ws[i] = 0.0f;
}

// ---------------------------------------------------------------------------
extern "C" void kernel_launch(void* const* d_in, const int* in_sizes, int n_in,
                              void* d_out, int out_size, void* d_ws, size_t ws_size,
                              hipStream_t stream) {
    const float* ur = (const float*)d_in[0];
    const float* ui = (const float*)d_in[1];
    float* out = (float*)d_out;

    float* strips = (float*)d_ws;                 // NBATCH * NSEG = 3200 floats
    float* sq     = strips + NBATCH * NSEG;       // NBATCH * NSQ  =  320 floats
    const int ws_elems = NBATCH * (NSEG + NSQ);   // 3520

    zero_ws_kernel<<<dim3((ws_elems + 255) / 256), dim3(256), 0, stream>>>(strips, ws_elems);

    // Main streaming pass: 128 col-groups (x=32 blocks of 4 waves),
    // 8 row-splits (y), 32 batches (z).
    colsum_wmma_kernel<<<dim3(32, 8, NBATCH), dim3(128), 0, stream>>>(ur, ui, strips);

    sqsum_kernel<<<dim3(NSQ, NBATCH), dim3(256), 0, stream>>>(ur, ui, sq);

    finalize_kernel<<<dim3(NBATCH), dim3(128), 0, stream>>>(sq, strips, out);
}